// AddConvStateNet_90881507983899
// MI455X (gfx1250) — compile-verified
//
#include <hip/hip_runtime.h>

#define NTASK  200000
#define NDATA  100000
#define EDT    1000000
#define ETT    1000000
#define DDATA  5
#define DTASK  12
#define HDIM   64
#define NT_PAD 200064   /* 1563 * 128 */
#define NBLK   1563
#define LDW    65       /* padded LDS row stride for C staging */
#define LDK8   12       /* transposed-W K-stride, K=8  (+4 pad) */
#define LDK12  16       /* transposed-W K-stride, K=12 (+4 pad) */
#define LDK64  68       /* transposed-W K-stride, K=64 (+4 pad) */
#define LDK128 132      /* transposed-W K-stride, K=128(+4 pad) */

typedef float v2f __attribute__((ext_vector_type(2)));
typedef float v8f __attribute__((ext_vector_type(8)));

__device__ __forceinline__ v8f wmma4(v2f a, v2f b, v8f c) {
  // D = A(16x4 f32) * B(4x16 f32) + C(16x16 f32)
  return __builtin_amdgcn_wmma_f32_16x16x4_f32(false, a, false, b, (short)0, c,
                                               false, false);
}
__device__ __forceinline__ float leaky(float x) { return x > 0.f ? x : 0.01f * x; }
// contiguous {k,k+1} B-fragment from transposed LDS weights (single ds_load_b64)
__device__ __forceinline__ v2f bfrag(const float* sWt, int ldk, int n, int k) {
  return *(const v2f*)&sWt[n * ldk + k];
}

// ---------------------------------------------------------------- utilities
__global__ void k_zero(float* __restrict__ p, long long n) {
  long long i = (long long)blockIdx.x * blockDim.x + threadIdx.x;
  long long st = (long long)gridDim.x * blockDim.x;
  for (; i < n; i += st) p[i] = 0.f;
}

// scatter-add data features into task accumulator (DataTaskGraphConv agg)
__global__ void k_scatter_agg(const float* __restrict__ data_x,
                              const int* __restrict__ src,
                              const int* __restrict__ dst,
                              float* __restrict__ agg) {
  int e = blockIdx.x * blockDim.x + threadIdx.x;
  if (e >= EDT) return;
  long long s = src[e], d = dst[e];
#pragma unroll
  for (int k = 0; k < DDATA; ++k)
    unsafeAtomicAdd(&agg[d * DDATA + k], data_x[s * DDATA + k]);
}

// per-node edge degrees for the deferred b2 term
__global__ void k_degrees(const int* __restrict__ tsrc, const int* __restrict__ tdst,
                          float* __restrict__ degd, float* __restrict__ degr) {
  int e = blockIdx.x * blockDim.x + threadIdx.x;
  if (e >= ETT) return;
  unsafeAtomicAdd(&degd[tdst[e]], 1.f);
  unsafeAtomicAdd(&degr[tsrc[e]], 1.f);
}

// build combined EdgeConv weights: Wu = W1[:64]-W1[64:], Wv = W1[64:]
__global__ void k_combine_w(const float* __restrict__ depW1,
                            const float* __restrict__ revW1,
                            float* __restrict__ wc) {
  int i = blockIdx.x * blockDim.x + threadIdx.x;
  if (i >= HDIM * HDIM) return;
  float da = depW1[i], db = depW1[HDIM * HDIM + i];
  float ra = revW1[i], rb = revW1[HDIM * HDIM + i];
  wc[0 * HDIM * HDIM + i] = da - db;
  wc[1 * HDIM * HDIM + i] = db;
  wc[2 * HDIM * HDIM + i] = ra - rb;
  wc[3 * HDIM * HDIM + i] = rb;
}

// -------------------------------------------------- fused = leaky(LN(agg@Wrel + b + tasks@Wroot))
__global__ __launch_bounds__(256) void k_fused(
    const float* __restrict__ agg, const float* __restrict__ tasks,
    const float* __restrict__ Wrel, const float* __restrict__ brel,
    const float* __restrict__ Wroot, const float* __restrict__ lng,
    const float* __restrict__ lnb, float* __restrict__ fused) {
  __shared__ float sW1[HDIM * LDK8];    // transposed [n][k], K=8 (rows 5..7 = 0)
  __shared__ float sW2[HDIM * LDK12];   // transposed [n][k], K=12
  __shared__ float sRows[128 * LDW];
  int tid = threadIdx.x;
  for (int i = tid; i < 8 * HDIM; i += 256) {
    int k = i >> 6, n = i & 63;
    sW1[n * LDK8 + k] = (k < DDATA) ? Wrel[k * HDIM + n] : 0.f;
  }
  for (int i = tid; i < 12 * HDIM; i += 256) {
    int k = i >> 6, n = i & 63;
    sW2[n * LDK12 + k] = Wroot[i];
  }
  __syncthreads();

  int lane = tid & 31, wave = tid >> 5;
  int m = lane & 15, hi2 = (lane >> 4) << 1;
  long long row = (long long)blockIdx.x * 128 + wave * 16 + m;
  bool valid = row < NTASK;

  v2f aA[2], aT[3];
#pragma unroll
  for (int ks = 0; ks < 2; ++ks) {
    int k = ks * 4 + hi2;
    v2f f = {0.f, 0.f};
    if (valid) {  // stride-5 rows: keep guarded scalar loads
      if (k < DDATA) f.x = agg[row * DDATA + k];
      if (k + 1 < DDATA) f.y = agg[row * DDATA + k + 1];
    }
    aA[ks] = f;
  }
#pragma unroll
  for (int ks = 0; ks < 3; ++ks) {
    int k = ks * 4 + hi2;  // even -> 8B-aligned pair
    v2f f = {0.f, 0.f};
    if (valid) f = *(const v2f*)&tasks[row * DTASK + k];
    aT[ks] = f;
  }
#pragma unroll
  for (int ct = 0; ct < 4; ++ct) {
    v8f c = {0.f, 0.f, 0.f, 0.f, 0.f, 0.f, 0.f, 0.f};
    int n = ct * 16 + m;
#pragma unroll
    for (int ks = 0; ks < 2; ++ks)
      c = wmma4(aA[ks], bfrag(sW1, LDK8, n, ks * 4 + hi2), c);
#pragma unroll
    for (int ks = 0; ks < 3; ++ks)
      c = wmma4(aT[ks], bfrag(sW2, LDK12, n, ks * 4 + hi2), c);
#pragma unroll
    for (int g = 0; g < 8; ++g) {
      int r = wave * 16 + g + 8 * (lane >> 4);
      sRows[r * LDW + n] = c[g];
    }
  }
  __syncthreads();
  if (tid < 128) {
    long long grow = (long long)blockIdx.x * 128 + tid;
    if (grow < NTASK) {
      float s1 = 0.f, s2 = 0.f;
      for (int c2 = 0; c2 < HDIM; ++c2) {
        float x = sRows[tid * LDW + c2] + brel[c2];
        s1 += x; s2 += x * x;
      }
      float mean = s1 * (1.f / HDIM);
      float var = s2 * (1.f / HDIM) - mean * mean;
      float inv = rsqrtf(var + 1e-5f);
      for (int c2 = 0; c2 < HDIM; ++c2) {
        float x = sRows[tid * LDW + c2] + brel[c2];
        fused[grow * HDIM + c2] = leaky((x - mean) * inv * lng[c2] + lnb[c2]);
      }
    }
  }
}

// -------------------------------------------------- u/v projections: fused @ {Wu_dep, Wv_dep, Wu_rev, Wv_rev}
__global__ __launch_bounds__(256) void k_uv(
    const float* __restrict__ fused, const float* __restrict__ wc,
    float* __restrict__ ud, float* __restrict__ vd,
    float* __restrict__ ur, float* __restrict__ vr) {
  __shared__ float sW[4 * HDIM * LDK64];  // 4 transposed 64x64 matrices
  int tid = threadIdx.x;
  for (int i = tid; i < 4 * HDIM * HDIM; i += 256) {
    int mtx = i >> 12, k = (i >> 6) & 63, n = i & 63;
    sW[mtx * HDIM * LDK64 + n * LDK64 + k] = wc[i];
  }
  __syncthreads();
  int lane = tid & 31, wave = tid >> 5;
  int m = lane & 15, hi2 = (lane >> 4) << 1;
  long long row = (long long)blockIdx.x * 128 + wave * 16 + m;
  bool valid = row < NTASK;
  v2f a[16];
#pragma unroll
  for (int ks = 0; ks < 16; ++ks) {
    int k = ks * 4 + hi2;
    v2f f = {0.f, 0.f};
    if (valid) f = *(const v2f*)&fused[row * HDIM + k];
    a[ks] = f;
  }
  float* outs[4] = {ud, vd, ur, vr};
#pragma unroll
  for (int mtx = 0; mtx < 4; ++mtx) {
    const float* Ws = &sW[mtx * HDIM * LDK64];
    float* O = outs[mtx];
#pragma unroll
    for (int ct = 0; ct < 4; ++ct) {
      v8f c = {0.f, 0.f, 0.f, 0.f, 0.f, 0.f, 0.f, 0.f};
      int n = ct * 16 + m;
#pragma unroll
      for (int ks = 0; ks < 16; ++ks)
        c = wmma4(a[ks], bfrag(Ws, LDK64, n, ks * 4 + hi2), c);
#pragma unroll
      for (int g = 0; g < 8; ++g) {
        long long rg = (long long)blockIdx.x * 128 + wave * 16 + g + 8 * (lane >> 4);
        if (rg < NTASK) O[rg * HDIM + n] = c[g];
      }
    }
  }
}

// -------------------------------------------------- edge pass: h = leaky(u[dst]+v[src]+b1); acc[dst] += h
__global__ __launch_bounds__(256) void k_edges(
    const int* __restrict__ tsrc, const int* __restrict__ tdst,
    const float* __restrict__ ud, const float* __restrict__ vd,
    const float* __restrict__ ur, const float* __restrict__ vr,
    const float* __restrict__ b1d, const float* __restrict__ b1r,
    float* __restrict__ accd, float* __restrict__ accr) {
  long long wid = ((long long)blockIdx.x * blockDim.x + threadIdx.x) >> 5;
  int lane = threadIdx.x & 31;
  if (wid >= 2LL * ETT) return;
  if (wid < ETT) {
    int e = (int)wid;
    long long d = tdst[e], s = tsrc[e];
#pragma unroll
    for (int half = 0; half < 2; ++half) {
      int c2 = lane + 32 * half;
      float h = leaky(ud[d * HDIM + c2] + vd[s * HDIM + c2] + b1d[c2]);
      unsafeAtomicAdd(&accd[d * HDIM + c2], h);
    }
  } else {
    int e = (int)(wid - ETT);
    long long d = tsrc[e], s = tdst[e];  // flipped edges
#pragma unroll
    for (int half = 0; half < 2; ++half) {
      int c2 = lane + 32 * half;
      float h = leaky(ur[d * HDIM + c2] + vr[s * HDIM + c2] + b1r[c2]);
      unsafeAtomicAdd(&accr[d * HDIM + c2], h);
    }
  }
}

// -------------------------------------------------- out = leaky(LN(acc@W2 + deg*b2))
__global__ __launch_bounds__(256) void k_post(
    const float* __restrict__ acc, const float* __restrict__ W2,
    const float* __restrict__ b2, const float* __restrict__ lng,
    const float* __restrict__ lnb, const float* __restrict__ deg,
    float* __restrict__ outp) {
  __shared__ float sW[HDIM * LDK64];  // transposed 64x64
  __shared__ float sRows[128 * LDW];
  int tid = threadIdx.x;
  for (int i = tid; i < HDIM * HDIM; i += 256) {
    int k = i >> 6, n = i & 63;
    sW[n * LDK64 + k] = W2[i];
  }
  __syncthreads();
  int lane = tid & 31, wave = tid >> 5;
  int m = lane & 15, hi2 = (lane >> 4) << 1;
  long long row = (long long)blockIdx.x * 128 + wave * 16 + m;
  bool valid = row < NTASK;
  v2f a[16];
#pragma unroll
  for (int ks = 0; ks < 16; ++ks) {
    int k = ks * 4 + hi2;
    v2f f = {0.f, 0.f};
    if (valid) f = *(const v2f*)&acc[row * HDIM + k];
    a[ks] = f;
  }
#pragma unroll
  for (int ct = 0; ct < 4; ++ct) {
    v8f c = {0.f, 0.f, 0.f, 0.f, 0.f, 0.f, 0.f, 0.f};
    int n = ct * 16 + m;
#pragma unroll
    for (int ks = 0; ks < 16; ++ks)
      c = wmma4(a[ks], bfrag(sW, LDK64, n, ks * 4 + hi2), c);
#pragma unroll
    for (int g = 0; g < 8; ++g) {
      int r = wave * 16 + g + 8 * (lane >> 4);
      sRows[r * LDW + n] = c[g];
    }
  }
  __syncthreads();
  if (tid < 128) {
    long long grow = (long long)blockIdx.x * 128 + tid;
    if (grow < NTASK) {
      float dg = deg[grow];
      float s1 = 0.f, s2 = 0.f;
      for (int c2 = 0; c2 < HDIM; ++c2) {
        float x = sRows[tid * LDW + c2] + dg * b2[c2];
        s1 += x; s2 += x * x;
      }
      float mean = s1 * (1.f / HDIM);
      float var = s2 * (1.f / HDIM) - mean * mean;
      float inv = rsqrtf(var + 1e-5f);
      for (int c2 = 0; c2 < HDIM; ++c2) {
        float x = sRows[tid * LDW + c2] + dg * b2[c2];
        outp[grow * HDIM + c2] = leaky((x - mean) * inv * lng[c2] + lnb[c2]);
      }
    }
  }
}

// -------------------------------------------------- tf = leaky([dep|rev]@projW + b); colsum += tf; out[64:128]=tf[0]
__global__ __launch_bounds__(256) void k_proj(
    const float* __restrict__ dep, const float* __restrict__ rev,
    const float* __restrict__ Wp, const float* __restrict__ bp,
    float* __restrict__ colsum, float* __restrict__ outv) {
  __shared__ float sW[HDIM * LDK128];  // transposed 128x64 -> [n][k]
  __shared__ float sRed[HDIM];
  int tid = threadIdx.x;
  for (int i = tid; i < 128 * HDIM; i += 256) {
    int k = i >> 6, n = i & 63;
    sW[n * LDK128 + k] = Wp[i];
  }
  if (tid < HDIM) sRed[tid] = 0.f;
  __syncthreads();
  int lane = tid & 31, wave = tid >> 5;
  int m = lane & 15, hi2 = (lane >> 4) << 1;
  long long row = (long long)blockIdx.x * 128 + wave * 16 + m;
  bool valid = row < NTASK;
  v2f a[32];
#pragma unroll
  for (int ks = 0; ks < 32; ++ks) {
    int k = ks * 4 + hi2;  // pairs never straddle the dep/rev boundary (k even)
    v2f f = {0.f, 0.f};
    if (valid)
      f = (k < HDIM) ? *(const v2f*)&dep[row * HDIM + k]
                     : *(const v2f*)&rev[row * HDIM + (k - HDIM)];
    a[ks] = f;
  }
#pragma unroll
  for (int ct = 0; ct < 4; ++ct) {
    v8f c = {0.f, 0.f, 0.f, 0.f, 0.f, 0.f, 0.f, 0.f};
    int n = ct * 16 + m;
#pragma unroll
    for (int ks = 0; ks < 32; ++ks)
      c = wmma4(a[ks], bfrag(sW, LDK128, n, ks * 4 + hi2), c);
    float bias = bp[n];
    float part = 0.f;
#pragma unroll
    for (int g = 0; g < 8; ++g) {
      long long rg = (long long)blockIdx.x * 128 + wave * 16 + g + 8 * (lane >> 4);
      float y = leaky(c[g] + bias);
      if (rg < NTASK) part += y;
      if (rg == 0) outv[64 + n] = y;  // candidate = tf[0]
    }
    atomicAdd(&sRed[n], part);  // ds_add_f32
  }
  __syncthreads();
  if (tid < HDIM) unsafeAtomicAdd(&colsum[tid], sRed[tid]);
}

// -------------------------------------------------- final: global_state + device branch
__global__ void k_final(const float* __restrict__ colsum, const int* __restrict__ counts,
                        const float* __restrict__ devx, const float* __restrict__ timex,
                        const float* __restrict__ devW, const float* __restrict__ devb,
                        float* __restrict__ outv) {
  int c = threadIdx.x;
  if (c >= HDIM) return;
  int c0i = counts[0] > 1 ? counts[0] : 1;
  outv[c] = colsum[c] / (float)c0i;
  float s = devb[c];
  for (int j = 0; j < 96; ++j) s += devx[j] * devW[j * HDIM + c];
  s += (timex[0] * (1.f / 100000.f)) * devW[96 * HDIM + c];
  outv[128 + c] = leaky(s);
}

extern "C" void kernel_launch(void* const* d_in, const int* in_sizes, int n_in,
                              void* d_out, int out_size, void* d_ws, size_t ws_size,
                              hipStream_t stream) {
  const float* data_x    = (const float*)d_in[0];
  const float* tasks_x   = (const float*)d_in[1];
  const float* devices_x = (const float*)d_in[2];
  const float* time_x    = (const float*)d_in[3];
  const int*   dt_src    = (const int*)d_in[4];
  const int*   dt_dst    = (const int*)d_in[5];
  const int*   tt_src    = (const int*)d_in[6];
  const int*   tt_dst    = (const int*)d_in[7];
  const int*   counts    = (const int*)d_in[8];
  const float* gc_W_rel  = (const float*)d_in[9];
  const float* gc_b_rel  = (const float*)d_in[10];
  const float* gc_W_root = (const float*)d_in[11];
  const float* gc_ln_g   = (const float*)d_in[12];
  const float* gc_ln_b   = (const float*)d_in[13];
  const float* dep_W1    = (const float*)d_in[14];
  const float* dep_b1    = (const float*)d_in[15];
  const float* dep_W2    = (const float*)d_in[16];
  const float* dep_b2    = (const float*)d_in[17];
  const float* dep_ln_g  = (const float*)d_in[18];
  const float* dep_ln_b  = (const float*)d_in[19];
  const float* rev_W1    = (const float*)d_in[20];
  const float* rev_b1    = (const float*)d_in[21];
  const float* rev_W2    = (const float*)d_in[22];
  const float* rev_b2    = (const float*)d_in[23];
  const float* rev_ln_g  = (const float*)d_in[24];
  const float* rev_ln_b  = (const float*)d_in[25];
  const float* dev_W     = (const float*)d_in[26];
  const float* dev_b     = (const float*)d_in[27];
  const float* proj_W    = (const float*)d_in[28];
  const float* proj_b    = (const float*)d_in[29];
  float* out = (float*)d_out;
  float* ws  = (float*)d_ws;

  const long long SLOT = (long long)NT_PAD * HDIM;
  float* fused = ws + 0 * SLOT;  // reused as acc_dep after k_uv
  float* udep  = ws + 1 * SLOT;  // reused as dep output
  float* vdep  = ws + 2 * SLOT;  // reused as rev output
  float* urev  = ws + 3 * SLOT;
  float* vrev  = ws + 4 * SLOT;
  float* accr  = ws + 5 * SLOT;
  float* accd  = fused;
  float* depo  = udep;
  float* revo  = vdep;
  float* agg   = ws + 6 * SLOT;                       // NT_PAD * DDATA
  float* degd  = agg + (long long)NT_PAD * DDATA;     // NT_PAD
  float* degr  = degd + NT_PAD;                       // NT_PAD
  float* wc    = degr + NT_PAD;                       // 4 * 64 * 64
  float* csum  = wc + 4 * HDIM * HDIM;                // 64

  long long zlen = (long long)NT_PAD * DDATA + 2LL * NT_PAD + 4 * HDIM * HDIM + HDIM;
  hipLaunchKernelGGL(k_zero, dim3(2048), dim3(256), 0, stream, agg, zlen);
  hipLaunchKernelGGL(k_zero, dim3(2048), dim3(256), 0, stream, accr, SLOT);
  hipLaunchKernelGGL(k_scatter_agg, dim3((EDT + 255) / 256), dim3(256), 0, stream,
                     data_x, dt_src, dt_dst, agg);
  hipLaunchKernelGGL(k_degrees, dim3((ETT + 255) / 256), dim3(256), 0, stream,
                     tt_src, tt_dst, degd, degr);
  hipLaunchKernelGGL(k_combine_w, dim3((HDIM * HDIM + 255) / 256), dim3(256), 0, stream,
                     dep_W1, rev_W1, wc);
  hipLaunchKernelGGL(k_fused, dim3(NBLK), dim3(256), 0, stream,
                     agg, tasks_x, gc_W_rel, gc_b_rel, gc_W_root, gc_ln_g, gc_ln_b, fused);
  hipLaunchKernelGGL(k_uv, dim3(NBLK), dim3(256), 0, stream,
                     fused, wc, udep, vdep, urev, vrev);
  hipLaunchKernelGGL(k_zero, dim3(2048), dim3(256), 0, stream, accd, SLOT);
  unsigned eblocks = (unsigned)((2LL * ETT * 32 + 255) / 256);
  hipLaunchKernelGGL(k_edges, dim3(eblocks), dim3(256), 0, stream,
                     tt_src, tt_dst, udep, vdep, urev, vrev, dep_b1, rev_b1, accd, accr);
  hipLaunchKernelGGL(k_post, dim3(NBLK), dim3(256), 0, stream,
                     accd, dep_W2, dep_b2, dep_ln_g, dep_ln_b, degd, depo);
  hipLaunchKernelGGL(k_post, dim3(NBLK), dim3(256), 0, stream,
                     accr, rev_W2, rev_b2, rev_ln_g, rev_ln_b, degr, revo);
  hipLaunchKernelGGL(k_proj, dim3(NBLK), dim3(256), 0, stream,
                     depo, revo, proj_W, proj_b, csum, out);
  hipLaunchKernelGGL(k_final, dim3(1), dim3(64), 0, stream,
                     csum, counts, devices_x, time_x, dev_W, dev_b, out);
}